// SensitivityBasedQuantizedLinearINT4_63917703299740
// MI455X (gfx1250) — compile-verified
//
#include <hip/hip_runtime.h>
#include <hip/hip_bf16.h>
#include <math.h>

// ---------------------------------------------------------------------------
// SensitivityBasedQuantizedLinear: 2D k-means (k=16) quantization of weights
// by (w, g^2) features, then out = X @ Wq^T with fp32 WMMA (16x16x4) and
// async global->LDS double-buffered staging (CDNA5 ASYNCcnt path).
// ---------------------------------------------------------------------------

typedef float v2f __attribute__((ext_vector_type(2)));
typedef float v8f __attribute__((ext_vector_type(8)));

#define NCLUST 16
#define KM_ITERS 5
#define RED_BLOCKS 512   // fixed block count -> deterministic partial layout

#if defined(__has_builtin)
#if __has_builtin(__builtin_amdgcn_s_wait_asynccnt)
#define WAIT_ASYNC(n) __builtin_amdgcn_s_wait_asynccnt(n)
#endif
#endif
#ifndef WAIT_ASYNC
#define WAIT_ASYNC(n) asm volatile("s_wait_asynccnt %0" :: "i"(n) : "memory")
#endif

// async copy of 16 bytes global -> LDS (per lane), tracked by ASYNCcnt
__device__ __forceinline__ void async_copy_b128(void* lds_ptr, const void* gptr) {
    unsigned lds_off = (unsigned)(uintptr_t)lds_ptr;   // low 32 bits = LDS offset
    asm volatile("global_load_async_to_lds_b128 %0, %1, off"
                 :: "v"(lds_off), "v"(gptr) : "memory");
}

// order-preserving float -> uint key
__device__ __forceinline__ unsigned mapkey(float f) {
    unsigned u = __float_as_uint(f);
    return (u & 0x80000000u) ? ~u : (u | 0x80000000u);
}
__device__ __forceinline__ float unkey(unsigned k) {
    unsigned u = (k & 0x80000000u) ? (k & 0x7FFFFFFFu) : ~k;
    return __uint_as_float(u);
}

// --------------------------- workspace init --------------------------------
__global__ void ws_init_kernel(unsigned* prefix, unsigned* counts, unsigned* minidx) {
    int t = threadIdx.x;
    if (t < NCLUST) { prefix[t] = 0u; counts[t] = 0u; minidx[t] = 0xFFFFFFFFu; }
}

// ------------------- rank selection (bitwise radix select) -----------------
__global__ __launch_bounds__(256) void rank_count_kernel(
    const float* __restrict__ Wt, int N,
    const unsigned* __restrict__ prefix, unsigned* __restrict__ counts, int bit) {
    __shared__ unsigned scnt[NCLUST];
    if (threadIdx.x < NCLUST) scnt[threadIdx.x] = 0u;

    unsigned cand[NCLUST];
    const unsigned bmask = 1u << bit;
#pragma unroll
    for (int r = 0; r < NCLUST; ++r) cand[r] = prefix[r] | bmask;

    unsigned cnt[NCLUST];
#pragma unroll
    for (int r = 0; r < NCLUST; ++r) cnt[r] = 0u;

    const int stride = gridDim.x * blockDim.x;
    for (int i = blockIdx.x * blockDim.x + threadIdx.x; i < N; i += stride) {
        unsigned key = mapkey(Wt[i]);
#pragma unroll
        for (int r = 0; r < NCLUST; ++r) cnt[r] += (key < cand[r]) ? 1u : 0u;
    }
    // wave32 shuffle tree reduce (deterministic)
#pragma unroll
    for (int r = 0; r < NCLUST; ++r)
        for (int off = 16; off > 0; off >>= 1)
            cnt[r] += __shfl_down((int)cnt[r], off, 32);
    __syncthreads();
    if ((threadIdx.x & 31) == 0) {
#pragma unroll
        for (int r = 0; r < NCLUST; ++r) atomicAdd(&scnt[r], cnt[r]);
    }
    __syncthreads();
    if (threadIdx.x < NCLUST) atomicAdd(&counts[threadIdx.x], scnt[threadIdx.x]);
}

__global__ void rank_update_kernel(unsigned* prefix, unsigned* counts, int bit, unsigned N) {
    int r = threadIdx.x;
    if (r < NCLUST) {
        unsigned long long rank = ((unsigned long long)r * (unsigned long long)(N - 1u)) / 15ull;
        if ((unsigned long long)counts[r] <= rank) prefix[r] |= (1u << bit);
        counts[r] = 0u;   // ready for next bit pass
    }
}

__global__ __launch_bounds__(256) void find_index_kernel(
    const float* __restrict__ Wt, int N,
    const unsigned* __restrict__ prefix, unsigned* __restrict__ minidx) {
    unsigned key_r[NCLUST];
#pragma unroll
    for (int r = 0; r < NCLUST; ++r) key_r[r] = prefix[r];
    const int stride = gridDim.x * blockDim.x;
    for (int i = blockIdx.x * blockDim.x + threadIdx.x; i < N; i += stride) {
        unsigned key = mapkey(Wt[i]);
#pragma unroll
        for (int r = 0; r < NCLUST; ++r)
            if (key == key_r[r]) atomicMin(&minidx[r], (unsigned)i);
    }
}

__global__ void init_cent_kernel(const float* __restrict__ G, int N,
                                 const unsigned* __restrict__ prefix,
                                 const unsigned* __restrict__ minidx,
                                 float* centw, float* cents) {
    int r = threadIdx.x;
    if (r < NCLUST) {
        float w = unkey(prefix[r]);
        unsigned idx = minidx[r];
        float g = (idx < (unsigned)N) ? G[idx] : 0.0f;
        centw[r] = w;
        cents[r] = g * g;
    }
}

// ------------------------------- k-means -----------------------------------
__global__ __launch_bounds__(256) void assign_partial_kernel(
    const float* __restrict__ Wt, const float* __restrict__ G, int N,
    const float* __restrict__ centw, const float* __restrict__ cents,
    float* __restrict__ psums /* [RED_BLOCKS][48] */) {
    float cw[NCLUST], cs[NCLUST];
#pragma unroll
    for (int r = 0; r < NCLUST; ++r) { cw[r] = centw[r]; cs[r] = cents[r]; }

    float sw[NCLUST], ss[NCLUST], sc[NCLUST];
#pragma unroll
    for (int r = 0; r < NCLUST; ++r) { sw[r] = 0.f; ss[r] = 0.f; sc[r] = 0.f; }

    const int stride = gridDim.x * blockDim.x;
    for (int i = blockIdx.x * blockDim.x + threadIdx.x; i < N; i += stride) {
        float w = Wt[i];
        float g = G[i];
        float s = g * g;
        int best = 0;
        float bd = 3.4e38f;
#pragma unroll
        for (int r = 0; r < NCLUST; ++r) {
            float dw = w - cw[r], ds = s - cs[r];
            float d = dw * dw + ds * ds;
            if (d < bd) { bd = d; best = r; }
        }
#pragma unroll
        for (int r = 0; r < NCLUST; ++r) {
            bool m = (best == r);
            sw[r] += m ? w : 0.f;
            ss[r] += m ? s : 0.f;
            sc[r] += m ? 1.f : 0.f;
        }
    }
    // deterministic wave reduce
#pragma unroll
    for (int r = 0; r < NCLUST; ++r)
        for (int off = 16; off > 0; off >>= 1) {
            sw[r] += __shfl_down(sw[r], off, 32);
            ss[r] += __shfl_down(ss[r], off, 32);
            sc[r] += __shfl_down(sc[r], off, 32);
        }
    __shared__ float red[8][48];
    int lane = threadIdx.x & 31, wid = threadIdx.x >> 5;
    if (lane == 0) {
#pragma unroll
        for (int r = 0; r < NCLUST; ++r) {
            red[wid][r] = sw[r];
            red[wid][16 + r] = ss[r];
            red[wid][32 + r] = sc[r];
        }
    }
    __syncthreads();
    if (threadIdx.x < 48) {
        float t = 0.f;
#pragma unroll
        for (int w8 = 0; w8 < 8; ++w8) t += red[w8][threadIdx.x];
        psums[(size_t)blockIdx.x * 48 + threadIdx.x] = t;
    }
}

__global__ void update_cent_kernel(const float* __restrict__ psums, int nblk,
                                   float* centw, float* cents) {
    __shared__ float tot[48];
    int t = threadIdx.x;
    if (t < 48) {
        float s = 0.f;
        for (int b = 0; b < nblk; ++b) s += psums[(size_t)b * 48 + t];
        tot[t] = s;
    }
    __syncthreads();
    if (t < NCLUST) {
        float cnt = tot[32 + t];
        float denom = fmaxf(cnt, 1.0f);
        if (cnt > 0.0f) {                      // empty clusters keep old centroid
            centw[t] = tot[t] / denom;
            cents[t] = tot[16 + t] / denom;
        }
    }
}

// final labels -> quantized weights (gather of centroid weight coordinate)
__global__ __launch_bounds__(256) void label_wq_kernel(
    const float* __restrict__ Wt, const float* __restrict__ G, int N,
    const float* __restrict__ centw, const float* __restrict__ cents,
    float* __restrict__ Wq) {
    float cw[NCLUST], cs[NCLUST];
#pragma unroll
    for (int r = 0; r < NCLUST; ++r) { cw[r] = centw[r]; cs[r] = cents[r]; }
    const int stride = gridDim.x * blockDim.x;
    for (int i = blockIdx.x * blockDim.x + threadIdx.x; i < N; i += stride) {
        float w = Wt[i];
        float g = G[i];
        float s = g * g;
        float bd = 3.4e38f, bw = cw[0];
#pragma unroll
        for (int r = 0; r < NCLUST; ++r) {
            float dw = w - cw[r], ds = s - cs[r];
            float d = dw * dw + ds * ds;
            if (d < bd) { bd = d; bw = cw[r]; }
        }
        Wq[i] = bw;
    }
}

// ------------------------- fp32 WMMA GEMM ----------------------------------
// Out[M,O] = X[M,K] * Wq[O,K]^T ; 128x128 block tile, BK=32, 8 waves,
// wave tile 64(M) x 32(N) -> 4x2 accumulators of v_wmma_f32_16x16x4_f32.
// Double-buffered LDS fed by global_load_async_to_lds_b128 (ASYNCcnt).
#define BM 128
#define BN 128
#define BK 32
#define LDSS 36   // padded stride (floats); keeps 16B alignment, avoids bank conflicts

__global__ __launch_bounds__(256) void gemm_wmma_f32_kernel(
    const float* __restrict__ X, const float* __restrict__ Wq,
    float* __restrict__ Out, int M, int K, int O) {
    __shared__ float Xs[2][BM * LDSS];
    __shared__ float Ws[2][BN * LDSS];

    const int tid = threadIdx.x;
    const int lane = tid & 31;
    const int wid = tid >> 5;
    const int waveM = wid >> 2;    // 0..1  -> 64-row slab
    const int waveN = wid & 3;     // 0..3  -> 32-col slab
    const int halfK = lane >> 4;   // K-pair select per WMMA layout
    const int lrow = lane & 15;

    const int m0 = blockIdx.y * BM;
    const int n0 = blockIdx.x * BN;

    v8f acc[4][2];
    const v8f vzero = {0.f, 0.f, 0.f, 0.f, 0.f, 0.f, 0.f, 0.f};
#pragma unroll
    for (int i = 0; i < 4; ++i)
#pragma unroll
        for (int j = 0; j < 2; ++j) acc[i][j] = vzero;

    // issue one BK stage of async copies (8 b128 per thread-iteration group)
    auto stage = [&](int buf, int kb) {
#pragma unroll
        for (int c = 0; c < 4; ++c) {
            int linear = tid + 256 * c;        // 0..1023
            int row = linear >> 3;             // 0..127
            int col4 = (linear & 7) << 2;      // 0,4,...,28
            async_copy_b128(&Xs[buf][row * LDSS + col4],
                            X + (size_t)(m0 + row) * K + kb + col4);
            async_copy_b128(&Ws[buf][row * LDSS + col4],
                            Wq + (size_t)(n0 + row) * K + kb + col4);
        }
    };

    const int nstage = K / BK;
    stage(0, 0);

    for (int s = 0; s < nstage; ++s) {
        const int cur = s & 1;
        if (s + 1 < nstage) {
            stage(cur ^ 1, (s + 1) * BK);  // overlap next stage with compute
            WAIT_ASYNC(8);                 // previous stage's 8 copies done
        } else {
            WAIT_ASYNC(0);
        }
        __syncthreads();

        const float* xs = &Xs[cur][0];
        const float* ws = &Ws[cur][0];
#pragma unroll
        for (int kk = 0; kk < BK; kk += 4) {
            const int kf = kk + 2 * halfK;
            v2f a[4], b[2];
#pragma unroll
            for (int i = 0; i < 4; ++i) {
                int mr = waveM * 64 + i * 16 + lrow;
                a[i].x = xs[mr * LDSS + kf];
                a[i].y = xs[mr * LDSS + kf + 1];
            }
#pragma unroll
            for (int j = 0; j < 2; ++j) {
                int nr = waveN * 32 + j * 16 + lrow;
                b[j].x = ws[nr * LDSS + kf];
                b[j].y = ws[nr * LDSS + kf + 1];
            }
#pragma unroll
            for (int i = 0; i < 4; ++i)
#pragma unroll
                for (int j = 0; j < 2; ++j)
                    acc[i][j] = __builtin_amdgcn_wmma_f32_16x16x4_f32(
                        false, a[i], false, b[j], (short)0, acc[i][j], false, false);
        }
        __syncthreads();   // all waves done reading buf before it is refilled
    }

    // epilogue: C/D layout -> VGPR v holds row v + 8*halfK, col = lane&15
#pragma unroll
    for (int i = 0; i < 4; ++i)
#pragma unroll
        for (int j = 0; j < 2; ++j)
#pragma unroll
            for (int v = 0; v < 8; ++v) {
                int m = m0 + waveM * 64 + i * 16 + v + 8 * halfK;
                int n = n0 + waveN * 32 + j * 16 + lrow;
                Out[(size_t)m * O + n] = acc[i][j][v];
            }
}

// ---------------------------------------------------------------------------
extern "C" void kernel_launch(void* const* d_in, const int* in_sizes, int n_in,
                              void* d_out, int out_size, void* d_ws, size_t ws_size,
                              hipStream_t stream) {
    const float* X  = (const float*)d_in[0];   // [B,S,K] fp32
    const float* Wt = (const float*)d_in[1];   // [O,K]   fp32
    const float* G  = (const float*)d_in[2];   // [O,K]   fp32
    float* Out = (float*)d_out;

    const int NW = in_sizes[1];                       // O*K
    int K = (int)llround(sqrt((double)NW));           // square weight (2048)
    const int O = NW / K;
    const int M = in_sizes[0] / K;                    // B*S (8192)

    // workspace layout
    char* ws = (char*)d_ws;
    float* Wq = (float*)ws;
    size_t off = (size_t)NW * sizeof(float);
    unsigned* prefix = (unsigned*)(ws + off); off += NCLUST * sizeof(unsigned);
    unsigned* counts = (unsigned*)(ws + off); off += NCLUST * sizeof(unsigned);
    unsigned* minidx = (unsigned*)(ws + off); off += NCLUST * sizeof(unsigned);
    float* centw = (float*)(ws + off); off += NCLUST * sizeof(float);
    float* cents = (float*)(ws + off); off += NCLUST * sizeof(float);
    float* psums = (float*)(ws + off); off += (size_t)RED_BLOCKS * 48 * sizeof(float);

    // 1) init
    ws_init_kernel<<<1, 64, 0, stream>>>(prefix, counts, minidx);

    // 2) bitwise radix select of the 16 evenly spaced order statistics of w
    for (int bit = 31; bit >= 0; --bit) {
        rank_count_kernel<<<RED_BLOCKS, 256, 0, stream>>>(Wt, NW, prefix, counts, bit);
        rank_update_kernel<<<1, NCLUST, 0, stream>>>(prefix, counts, bit, (unsigned)NW);
    }
    find_index_kernel<<<RED_BLOCKS, 256, 0, stream>>>(Wt, NW, prefix, minidx);
    init_cent_kernel<<<1, NCLUST, 0, stream>>>(G, NW, prefix, minidx, centw, cents);

    // 3) Lloyd iterations
    for (int it = 0; it < KM_ITERS; ++it) {
        assign_partial_kernel<<<RED_BLOCKS, 256, 0, stream>>>(Wt, G, NW, centw, cents, psums);
        update_cent_kernel<<<1, 64, 0, stream>>>(psums, RED_BLOCKS, centw, cents);
    }

    // 4) labels -> quantized weights
    label_wq_kernel<<<RED_BLOCKS, 256, 0, stream>>>(Wt, G, NW, centw, cents, Wq);

    // 5) Out = X @ Wq^T via fp32 WMMA (async double-buffered)
    dim3 grid(O / BN, M / BM);
    gemm_wmma_f32_kernel<<<grid, 256, 0, stream>>>(X, Wq, Out, M, K, O);
}